// ProtoSegNet_79164837200695
// MI455X (gfx1250) — compile-verified
//
#include <hip/hip_runtime.h>
#include <math.h>

typedef float v2f __attribute__((ext_vector_type(2)));
typedef float v8f __attribute__((ext_vector_type(8)));

#define C_DIM 768
#define HW    4096      // 64*64
#define NPIX  32768     // 8*64*64
#define KP    10        // prototypes per set
#define NS    12        // N_SAMPLES
#define MT    4         // 16-pixel tiles per wave (64 pixels/wave)

__device__ inline float wred_sum(float v) {
    for (int o = 16; o > 0; o >>= 1) v += __shfl_xor(v, o, 32);
    return v;
}

// ---------------------------------------------------------------------------
// Kernel 1: per-pixel  max_k <F_p, proto_k>  (fg and bg) + ||F_p||, via
// V_WMMA_F32_16X16X4_F32.  One wave handles MT=4 tiles of 16 consecutive
// pixels (never crossing an image boundary since 4096 % 64 == 0), reusing
// each prototype A-fragment across the 4 pixel B-fragments.
//   A (16x4)  : protos, lane l<16 holds row l at c0+{0,1}; l>=16 row l-16 at c0+{2,3}
//   B (4x16)  : pixels, lane l<16 holds pixel l at K=c0,c0+1 ; l>=16 at c0+2,c0+3
//   D (16x16) : vgpr v, lanes<16 -> (M=v, N=lane); lanes>=16 -> (M=v+8)
// ---------------------------------------------------------------------------
__global__ __launch_bounds__(256)
void score_kernel(const float* __restrict__ fg, const float* __restrict__ bg,
                  const float* __restrict__ F,  const float* __restrict__ M,
                  float* __restrict__ score_fg, float* __restrict__ score_bg,
                  float* __restrict__ inv_norm)
{
    const int lane = threadIdx.x & 31;
    const int wave = threadIdx.x >> 5;
    const int wid  = blockIdx.x * 8 + wave;       // 0..511
    const int pix_base = wid * (16 * MT);
    const int n   = pix_base >> 12;               // image index
    const int hw  = pix_base & 4095;
    const int half = lane >> 4;                   // 0 or 1
    const int l16  = lane & 15;

    // A pointers (clamp rows 10..15 in-bounds; their results are never reduced)
    const int prow = (l16 < KP) ? l16 : (KP - 1);
    const float* fgp = fg + prow * C_DIM + 2 * half;
    const float* bgp = bg + prow * C_DIM + 2 * half;

    // B pointer: F[(n*768 + c)*4096 + hw + l16], c = c0 + 2*half (+1)
    const float* Fb = F + (size_t)(n * C_DIM) * HW + (size_t)(2 * half) * HW
                        + hw + l16;

    v8f dfg[MT], dbg[MT];
    float nrm[MT];
    #pragma unroll
    for (int t = 0; t < MT; ++t) {
        dfg[t] = (v8f){0.f, 0.f, 0.f, 0.f, 0.f, 0.f, 0.f, 0.f};
        dbg[t] = (v8f){0.f, 0.f, 0.f, 0.f, 0.f, 0.f, 0.f, 0.f};
        nrm[t] = 0.f;
    }

    for (int c0 = 0; c0 < C_DIM; c0 += 4) {
        v2f a_fg = *(const v2f*)(fgp + c0);
        v2f a_bg = *(const v2f*)(bgp + c0);
        #pragma unroll
        for (int t = 0; t < MT; ++t) {
            float bx = Fb[(size_t)c0 * HW + t * 16];
            float by = Fb[(size_t)(c0 + 1) * HW + t * 16];
            v2f b = {bx, by};
            nrm[t] = fmaf(bx, bx, nrm[t]);
            nrm[t] = fmaf(by, by, nrm[t]);
            dfg[t] = __builtin_amdgcn_wmma_f32_16x16x4_f32(false, a_fg, false, b,
                                                           (short)0, dfg[t], false, false);
            dbg[t] = __builtin_amdgcn_wmma_f32_16x16x4_f32(false, a_bg, false, b,
                                                           (short)0, dbg[t], false, false);
        }
    }

    #pragma unroll
    for (int t = 0; t < MT; ++t) {
        // max over protos 0..9: lanes<16 own M=0..7 (8 vgprs), lanes>=16 own M=8..9
        float f8 = dfg[t][0]; float b8 = dbg[t][0];
        #pragma unroll
        for (int i = 1; i < 8; ++i) {
            f8 = fmaxf(f8, dfg[t][i]);
            b8 = fmaxf(b8, dbg[t][i]);
        }
        float f2 = fmaxf(dfg[t][0], dfg[t][1]);
        float b2 = fmaxf(dbg[t][0], dbg[t][1]);
        float mfg = (half == 0) ? f8 : f2;
        float mbg = (half == 0) ? b8 : b2;
        mfg = fmaxf(mfg, __shfl_xor(mfg, 16, 32));
        mbg = fmaxf(mbg, __shfl_xor(mbg, 16, 32));

        float nt  = nrm[t] + __shfl_xor(nrm[t], 16, 32);
        float inv = 1.0f / fmaxf(sqrtf(nt), 1e-8f);

        if (half == 0) {
            int p = pix_base + t * 16 + l16;
            float m = M[p];
            m = fminf(fmaxf(m, 0.f), 1.f);
            score_fg[p] = (1.f - mfg * inv) * m;
            score_bg[p] = (1.f - mbg * inv) * (1.f - m);
            inv_norm[p] = inv;
        }
    }
}

// ---------------------------------------------------------------------------
// Kernel 2: top-12 indices (descending, ties -> smaller index, like lax.top_k).
// block 0 -> fg, block 1 -> bg.  Exclusion list kept in LDS (no global RAW).
// ---------------------------------------------------------------------------
__global__ __launch_bounds__(1024)
void topk12_kernel(const float* __restrict__ sfg, const float* __restrict__ sbg,
                   int* __restrict__ pos_idx, int* __restrict__ neg_idx)
{
    const float* s = (blockIdx.x == 0) ? sfg : sbg;
    int* out       = (blockIdx.x == 0) ? pos_idx : neg_idx;
    __shared__ float vals[1024];
    __shared__ int   ids[1024];
    __shared__ int   sel[NS];
    const int tid = threadIdx.x;

    for (int r = 0; r < NS; ++r) {
        float bv = -INFINITY; int bi = 0x7fffffff;
        for (int j = tid; j < NPIX; j += 1024) {
            bool taken = false;
            for (int q = 0; q < r; ++q) taken |= (sel[q] == j);
            float v = taken ? -INFINITY : s[j];
            if (v > bv) { bv = v; bi = j; }       // increasing j => ties keep smallest
        }
        vals[tid] = bv; ids[tid] = bi;
        __syncthreads();
        for (int off = 512; off > 0; off >>= 1) {
            if (tid < off) {
                float ov = vals[tid + off]; int oi = ids[tid + off];
                if (ov > vals[tid] || (ov == vals[tid] && oi < ids[tid])) {
                    vals[tid] = ov; ids[tid] = oi;
                }
            }
            __syncthreads();
        }
        if (tid == 0) { sel[r] = ids[0]; out[r] = ids[0]; }
        __syncthreads();
    }
}

// ---------------------------------------------------------------------------
// Kernel 3: gather + normalize the 24 selected feature rows into workspace.
// ---------------------------------------------------------------------------
__global__ __launch_bounds__(256)
void gather_kernel(const float* __restrict__ F, const float* __restrict__ inv_norm,
                   const int* __restrict__ pos_idx, const int* __restrict__ neg_idx,
                   float* __restrict__ pos_feats, float* __restrict__ neg_feats)
{
    const int b   = blockIdx.x;                 // 0..23
    const int row = (b < NS) ? b : b - NS;
    const int p   = (b < NS) ? pos_idx[row] : neg_idx[row];
    float* dst    = ((b < NS) ? pos_feats : neg_feats) + row * C_DIM;
    const int n = p >> 12, hw = p & 4095;
    const float* src = F + (size_t)(n * C_DIM) * HW + hw;
    const float inv = inv_norm[p];
    for (int c = threadIdx.x; c < C_DIM; c += 256)
        dst[c] = src[(size_t)c * HW] * inv;
}

// ---------------------------------------------------------------------------
// Kernel 4: 10-iteration spherical k-means refinement (fg then bg),
// triplet + InfoNCE loss, blended re-normalized prototype outputs.
// Single block; evolving state lives in LDS (P, S ~ 60 KB).
// ---------------------------------------------------------------------------
__global__ __launch_bounds__(256)
void refine_loss_kernel(const float* __restrict__ fg, const float* __restrict__ bg,
                        const float* __restrict__ pos_feats,
                        const float* __restrict__ neg_feats,
                        float* __restrict__ out)
{
    __shared__ float P[KP * C_DIM];       // current prototypes
    __shared__ float S[KP * C_DIM];       // segment sums / candidates / scratch
    __shared__ float Spp[NS * KP], Snp[NS * KP], Spb[NS * KP];
    __shared__ float simbuf[NS * KP];
    __shared__ int   assign_[NS];
    __shared__ float cnts[16];
    __shared__ float norms[16];

    const int tid  = threadIdx.x;
    const int lane = tid & 31;
    const int wave = tid >> 5;

    for (int phase = 0; phase < 2; ++phase) {
        const float* proto = phase ? bg : fg;
        const float* feats = phase ? neg_feats : pos_feats;

        for (int i = tid; i < KP * C_DIM; i += 256) P[i] = proto[i];
        __syncthreads();

        for (int it = 0; it < 10; ++it) {
            const float step = 0.1f / (1.0f + 0.5f * (float)it);

            // sims[s][k] = <feats_s, P_k>, one wave per (s,k) pair
            for (int pr = wave; pr < NS * KP; pr += 8) {
                const int s = pr / KP, k = pr % KP;
                const float* fs = feats + s * C_DIM;
                const float* pk = P + k * C_DIM;
                float acc = 0.f;
                for (int c = lane; c < C_DIM; c += 32) acc = fmaf(fs[c], pk[c], acc);
                acc = wred_sum(acc);
                if (lane == 0) simbuf[pr] = acc;
            }
            __syncthreads();

            if (tid == 0) {
                for (int k = 0; k < KP; ++k) cnts[k] = 0.f;
                for (int s = 0; s < NS; ++s) {
                    int best = 0; float bv = simbuf[s * KP];
                    for (int k = 1; k < KP; ++k) {
                        float v = simbuf[s * KP + k];
                        if (v > bv) { bv = v; best = k; }   // ties -> first (jnp.argmax)
                    }
                    assign_[s] = best;
                    cnts[best] += 1.f;
                }
            }
            __syncthreads();

            // segment sums into S (each thread owns distinct columns c)
            for (int c = tid; c < C_DIM; c += 256) {
                for (int k = 0; k < KP; ++k) S[k * C_DIM + c] = 0.f;
                for (int s = 0; s < NS; ++s)
                    S[assign_[s] * C_DIM + c] += feats[s * C_DIM + c];
            }
            __syncthreads();

            // candidate: (1-step)*p + step * sums/max(cnt,1)   (in-place in S)
            for (int c = tid; c < C_DIM; c += 256)
                for (int k = 0; k < KP; ++k) {
                    float mean = S[k * C_DIM + c] / fmaxf(cnts[k], 1.0f);
                    S[k * C_DIM + c] = (1.0f - step) * P[k * C_DIM + c] + step * mean;
                }
            __syncthreads();

            for (int k = wave; k < KP; k += 8) {
                float ss = 0.f;
                for (int c = lane; c < C_DIM; c += 32) {
                    float v = S[k * C_DIM + c]; ss = fmaf(v, v, ss);
                }
                ss = wred_sum(ss);
                if (lane == 0) norms[k] = sqrtf(ss);
            }
            __syncthreads();

            for (int c = tid; c < C_DIM; c += 256)
                for (int k = 0; k < KP; ++k)
                    if (cnts[k] > 0.f)
                        P[k * C_DIM + c] = S[k * C_DIM + c] / fmaxf(norms[k], 1e-8f);
            __syncthreads();
        }

        // post-refine similarity matrices while P is live in LDS
        float* Sm1 = phase ? Spb : Spp;            // pos @ P
        for (int pr = wave; pr < NS * KP; pr += 8) {
            const int s = pr / KP, k = pr % KP;
            const float* fs = pos_feats + s * C_DIM;
            const float* pk = P + k * C_DIM;
            float acc = 0.f;
            for (int c = lane; c < C_DIM; c += 32) acc = fmaf(fs[c], pk[c], acc);
            acc = wred_sum(acc);
            if (lane == 0) Sm1[pr] = acc;
        }
        if (phase == 0) {                          // neg @ p_fg
            for (int pr = wave; pr < NS * KP; pr += 8) {
                const int s = pr / KP, k = pr % KP;
                const float* fs = neg_feats + s * C_DIM;
                const float* pk = P + k * C_DIM;
                float acc = 0.f;
                for (int c = lane; c < C_DIM; c += 32) acc = fmaf(fs[c], pk[c], acc);
                acc = wred_sum(acc);
                if (lane == 0) Snp[pr] = acc;
            }
        }
        __syncthreads();

        // refined = safe_norm(0.7*proto + 0.3*P)
        for (int c = tid; c < C_DIM; c += 256)
            for (int k = 0; k < KP; ++k)
                S[k * C_DIM + c] = 0.7f * proto[k * C_DIM + c] + 0.3f * P[k * C_DIM + c];
        __syncthreads();
        for (int k = wave; k < KP; k += 8) {
            float ss = 0.f;
            for (int c = lane; c < C_DIM; c += 32) {
                float v = S[k * C_DIM + c]; ss = fmaf(v, v, ss);
            }
            ss = wred_sum(ss);
            if (lane == 0) norms[k] = sqrtf(ss);
        }
        __syncthreads();
        float* dst = out + 1 + phase * (KP * C_DIM);
        for (int c = tid; c < C_DIM; c += 256)
            for (int k = 0; k < KP; ++k)
                dst[k * C_DIM + c] = S[k * C_DIM + c] / fmaxf(norms[k], 1e-8f);
        __syncthreads();
    }

    if (tid == 0) {
        float sp = 0.f, sn = 0.f;
        for (int s = 0; s < NS; ++s) {
            float m = Spp[s * KP];
            for (int k = 1; k < KP; ++k) m = fmaxf(m, Spp[s * KP + k]);
            sp += m;
        }
        for (int s = 0; s < NS; ++s) {
            float m = Snp[s * KP];
            for (int k = 1; k < KP; ++k) m = fmaxf(m, Snp[s * KP + k]);
            sn += m;
        }
        sp /= (float)NS; sn /= (float)NS;
        float loss = fmaxf(0.f, 0.2f + sn - sp);

        const float itau = 1.0f / 0.07f;
        float acc = 0.f;
        for (int s = 0; s < NS; ++s) {
            float m1 = -INFINITY;
            for (int k = 0; k < KP; ++k) m1 = fmaxf(m1, Spp[s * KP + k] * itau);
            float e1 = 0.f;
            for (int k = 0; k < KP; ++k) e1 += expf(Spp[s * KP + k] * itau - m1);
            float numer = m1 + logf(e1);
            float m2 = m1;
            for (int k = 0; k < KP; ++k) m2 = fmaxf(m2, Spb[s * KP + k] * itau);
            float e2 = 0.f;
            for (int k = 0; k < KP; ++k) e2 += expf(Spp[s * KP + k] * itau - m2);
            for (int k = 0; k < KP; ++k) e2 += expf(Spb[s * KP + k] * itau - m2);
            float denom = m2 + logf(e2);
            acc += (numer - denom);
        }
        loss += 0.25f * (-(acc / (float)NS));
        out[0] = loss;
    }
}

// ---------------------------------------------------------------------------
extern "C" void kernel_launch(void* const* d_in, const int* in_sizes, int n_in,
                              void* d_out, int out_size, void* d_ws, size_t ws_size,
                              hipStream_t stream)
{
    const float* fg = (const float*)d_in[0];   // [10,768]
    const float* bg = (const float*)d_in[1];   // [10,768]
    const float* F  = (const float*)d_in[2];   // [8,768,64,64]
    const float* M  = (const float*)d_in[3];   // [8,1,64,64]
    float* out = (float*)d_out;                // 1 + 7680 + 7680

    char* ws = (char*)d_ws;
    float* score_fg  = (float*)(ws);                    // 32768 f32
    float* score_bg  = (float*)(ws + 131072);           // 32768 f32
    float* inv_norm  = (float*)(ws + 262144);           // 32768 f32
    int*   pos_idx   = (int*)  (ws + 393216);           // 12 int (pad 64)
    int*   neg_idx   = (int*)  (ws + 393280);           // 12 int (pad 64)
    float* pos_feats = (float*)(ws + 393344);           // 12*768 f32
    float* neg_feats = (float*)(ws + 430208);           // 12*768 f32

    score_kernel<<<64, 256, 0, stream>>>(fg, bg, F, M, score_fg, score_bg, inv_norm);
    topk12_kernel<<<2, 1024, 0, stream>>>(score_fg, score_bg, pos_idx, neg_idx);
    gather_kernel<<<24, 256, 0, stream>>>(F, inv_norm, pos_idx, neg_idx, pos_feats, neg_feats);
    refine_loss_kernel<<<1, 256, 0, stream>>>(fg, bg, pos_feats, neg_feats, out);
}